// LSTMModel_66451734004100
// MI455X (gfx1250) — compile-verified
//
#include <hip/hip_runtime.h>
#include <hip/hip_bf16.h>
#include <math.h>

// Problem constants (match reference)
#define NMOL  32
#define NATOM 128
#define DEG   32
#define NN    (NMOL*NATOM)     // 4096 atoms
#define PP    (NN*DEG)         // 131072 pairs
#define NSENS 20
#define NF1   20
#define NF0   40
#define NG    80
#define NIN0  5                // 2 one-hot + 3 history
#define KMAX  800              // largest GEMM K (envW)

typedef float v2f __attribute__((ext_vector_type(2)));
typedef float v8f __attribute__((ext_vector_type(8)));

// gfx1250 async global->LDS copy path (ASYNCcnt), guarded for portability.
// Builtin signature (from hipcc diagnostic): (global int*, lds int*, imm, imm).
#if __has_builtin(__builtin_amdgcn_global_load_async_to_lds_b32) && \
    __has_builtin(__builtin_amdgcn_s_wait_asynccnt)
#define HAVE_ASYNC_LDS 1
typedef __attribute__((address_space(1))) int* g32_ptr;
typedef __attribute__((address_space(3))) int* l32_ptr;
#define ASYNC_G2L_B32(ldst, gsrc)                                             \
    __builtin_amdgcn_global_load_async_to_lds_b32(                            \
        (g32_ptr)(int*)(gsrc), (l32_ptr)(int*)(ldst), 0, 0)
#define ASYNC_WAIT() __builtin_amdgcn_s_wait_asynccnt(0)
#else
#define HAVE_ASYNC_LDS 0
#define ASYNC_G2L_B32(ldst, gsrc) (*(ldst) = *(gsrc))
#define ASYNC_WAIT() ((void)0)
#endif

__device__ __forceinline__ float softplus_f(float x) {
    // stable: max(x,0) + log1p(exp(-|x|))
    return fmaxf(x, 0.0f) + log1pf(expf(-fabsf(x)));
}
__device__ __forceinline__ float sigmoid_f(float x) {
    return 1.0f / (1.0f + expf(-x));
}

// ---------------------------------------------------------------------------
// Pair sensitivities: sense[p][s] = exp(-0.5*((1/d - mu_s)/sigma)^2) * cut(d)
// ---------------------------------------------------------------------------
__global__ void sense_kernel(const float* __restrict__ coord,
                             const int* __restrict__ pf,
                             const int* __restrict__ ps,
                             float* __restrict__ sense) {
    int p = blockIdx.x * blockDim.x + threadIdx.x;
    if (p >= PP) return;
    int a = pf[p], b = ps[p];
    float dx = coord[a*3+0] - coord[b*3+0];
    float dy = coord[a*3+1] - coord[b*3+1];
    float dz = coord[a*3+2] - coord[b*3+2];
    float d  = sqrtf(dx*dx + dy*dy + dz*dz + 1e-12f);
    const float lo    = 1.0f/5.0f;          // 1/SOFT_MAX
    const float hi    = 1.0f/0.7f;          // 1/SOFT_MIN
    const float stp   = (hi - lo) / (float)(NSENS - 1);
    const float sigma = (hi - lo) / (float)NSENS;
    float invd = 1.0f / d;
    float cut = 0.0f;
    if (d < 7.5f) {
        float c = cosf(3.14159265358979323846f * d / 15.0f);
        cut = c * c;
    }
    float* out = sense + (size_t)p * NSENS;
    #pragma unroll
    for (int s = 0; s < NSENS; ++s) {
        float z = (invd - (lo + stp * (float)s)) / sigma;
        out[s] = expf(-0.5f * z * z) * cut;
    }
}

// ---------------------------------------------------------------------------
// ind_f = [onehot(species in {1,8}), x_h]  -> (N, 5)
// ---------------------------------------------------------------------------
__global__ void indf_kernel(const int* __restrict__ species,
                            const float* __restrict__ x_h,
                            float* __restrict__ indf) {
    int i = blockIdx.x * blockDim.x + threadIdx.x;
    if (i >= NN) return;
    int sp = species[i];
    float* r = indf + (size_t)i * NIN0;
    r[0] = (sp == 1) ? 1.0f : 0.0f;
    r[1] = (sp == 8) ? 1.0f : 0.0f;
    r[2] = x_h[i*3+0];
    r[3] = x_h[i*3+1];
    r[4] = x_h[i*3+2];
}

// ---------------------------------------------------------------------------
// Fused message + segment_sum. pair_first = repeat(arange(N), DEG), so each
// atom's segment is its DEG contiguous pairs:
//   env[i][s][f] = sum_k sense[i*DEG+k][s] * x[ps[i*DEG+k]][f]
// One block per atom; sense slice + gathered neighbor rows staged in LDS via
// the gfx1250 async global->LDS path (ASYNCcnt) when available.
// ---------------------------------------------------------------------------
template<int F>
__global__ void env_kernel(const float* __restrict__ x, int xstride,
                           const int* __restrict__ ps,
                           const float* __restrict__ sense,
                           float* __restrict__ env) {
    __shared__ float s_s[DEG * NSENS];
    __shared__ float s_x[DEG * F];
    const int atom = blockIdx.x;
    const int tid  = threadIdx.x;
    const int p0   = atom * DEG;
    for (int t = tid; t < DEG * NSENS; t += blockDim.x)
        ASYNC_G2L_B32(s_s + t, sense + (size_t)p0 * NSENS + t);
    for (int t = tid; t < DEG * F; t += blockDim.x) {
        int k = t / F, f = t - k * F;
        int nb = ps[p0 + k];
        ASYNC_G2L_B32(s_x + t, x + (size_t)nb * xstride + f);
    }
    ASYNC_WAIT();
    __syncthreads();
    for (int t = tid; t < NSENS * F; t += blockDim.x) {
        int s = t / F, f = t - s * F;
        float acc = 0.0f;
        #pragma unroll
        for (int k = 0; k < DEG; ++k)
            acc = fmaf(s_s[k * NSENS + s], s_x[k * F + f], acc);
        env[(size_t)atom * (NSENS * F) + t] = acc;
    }
}

// ---------------------------------------------------------------------------
// Generic fp32 GEMM on the matrix pipes:
//   D[:, dcol:dcol+Nc] = act( A(M x K) @ W(K x Nc) + X(M x KX) @ S(KX x Nc) + bias )
// One 16x16 tile per wave via V_WMMA_F32_16X16X4_F32 (full fp32, K-step 4).
// The block's K x 16 B-slab (shared by all 4 m-tile waves) is staged once in
// LDS with async global->LDS copies; B fragments then come from ds_load.
// A fragment (16x4, §7.12.2): lane -> M = lane&15; VGPR0/1 hold K = 2*(lane>>4)+{0,1}
//   -> one contiguous float2 global load per lane per k-step.
// C/D: VGPR j -> row m0 + j + 8*(lane>>4), col n0 + (lane&15).
// M = 4096 (exact tiles); Nc edges masked on store, B col index clamped.
// ACT: 0 = identity, 1 = softplus.
// ---------------------------------------------------------------------------
template<int ACT>
__global__ void gemm_wmma(const float* __restrict__ A, int lda, int K,
                          const float* __restrict__ W, int ldw,
                          const float* __restrict__ X, int xstride, int KX,
                          const float* __restrict__ S,
                          const float* __restrict__ bias,
                          float* __restrict__ D, int ldd, int dcol,
                          int Nc) {
    __shared__ float sB[KMAX * 16];        // 51.2 KB of the 320 KB/WGP LDS
    const int lane = threadIdx.x;          // wave32: 0..31
    const int lo   = lane & 15;
    const int hi   = lane >> 4;
    const int m0   = (blockIdx.x * blockDim.y + threadIdx.y) * 16;
    const int n0   = blockIdx.y * 16;
    const int ncol = n0 + lo;
    const int tid  = threadIdx.y * 32 + threadIdx.x;   // 0..127

    // Stage B slab: sB[k*16 + c] = W[k][clamp(n0+c)]
    for (int t = tid; t < K * 16; t += 128) {
        int k = t >> 4, c = t & 15;
        int col = n0 + c; col = (col < Nc) ? col : (Nc - 1);
        ASYNC_G2L_B32(sB + t, W + (size_t)k * ldw + col);
    }
    ASYNC_WAIT();
    __syncthreads();

    v8f acc = {0.f,0.f,0.f,0.f,0.f,0.f,0.f,0.f};

#if __has_builtin(__builtin_amdgcn_wmma_f32_16x16x4_f32)
    const float* arow = A + (size_t)(m0 + lo) * lda + 2 * hi;
    const float* brow = sB + 2 * hi * 16 + lo;
    for (int k0 = 0; k0 < K; k0 += 4) {
        v2f a = *reinterpret_cast<const v2f*>(arow + k0);  // A[m][k0+2hi, +1]
        v2f b;
        b.x = brow[k0 * 16];           // sB[(k0+2hi)*16   + lo]
        b.y = brow[k0 * 16 + 16];      // sB[(k0+2hi+1)*16 + lo]
        acc = __builtin_amdgcn_wmma_f32_16x16x4_f32(
                  false, a, false, b, (short)0, acc, false, false);
    }
#else
    // Portable fallback with identical accumulator layout (host pass / probe).
    #pragma unroll
    for (int j = 0; j < 8; ++j) {
        int row = m0 + j + 8 * hi;
        float v = 0.0f;
        for (int k = 0; k < K; ++k)
            v = fmaf(A[(size_t)row * lda + k], sB[k * 16 + lo], v);
        acc[j] = v;
    }
#endif

    if (ncol < Nc) {
        float bval = bias ? bias[ncol] : 0.0f;
        #pragma unroll
        for (int j = 0; j < 8; ++j) {
            int row = m0 + j + 8 * hi;
            float v = acc[j] + bval;
            for (int kx = 0; kx < KX; ++kx)
                v = fmaf(X[(size_t)row * xstride + kx], S[(size_t)kx * Nc + ncol], v);
            if (ACT == 1) v = softplus_f(v);
            D[(size_t)row * ldd + dcol + ncol] = v;
        }
    }
}

// ---------------------------------------------------------------------------
// Split f0_out(N,40) -> c = f0[:, :20], cat[:, 20:40] = f0[:, 20:] (h state)
// ---------------------------------------------------------------------------
__global__ void split_kernel(const float* __restrict__ f0,
                             float* __restrict__ c,
                             float* __restrict__ cat) {
    int i = blockIdx.x * blockDim.x + threadIdx.x;
    if (i >= NN * NF1) return;
    int r = i / NF1, j = i - r * NF1;
    c[i] = f0[(size_t)r * NF0 + j];
    cat[(size_t)r * NF0 + NF1 + j] = f0[(size_t)r * NF0 + NF1 + j];
}

// ---------------------------------------------------------------------------
// LSTM gate update from pre-activations tmp(N,80) viewed as (N,20,4):
//   c = sig(t1)*c + sig(t0)*tanh(t2); o = sig(t3); h = o*tanh(c)
// h written straight into cat[:, 20:40].
// ---------------------------------------------------------------------------
__global__ void gate_kernel(const float* __restrict__ tmp,
                            float* __restrict__ c,
                            float* __restrict__ o,
                            float* __restrict__ cat) {
    int i = blockIdx.x * blockDim.x + threadIdx.x;
    if (i >= NN * NF1) return;
    int r = i / NF1, j = i - r * NF1;
    const float* t = tmp + (size_t)r * NG + j * 4;
    float cn = sigmoid_f(t[1]) * c[i] + sigmoid_f(t[0]) * tanhf(t[2]);
    float ov = sigmoid_f(t[3]);
    c[i] = cn;
    o[i] = ov;
    cat[(size_t)r * NF0 + NF1 + j] = ov * tanhf(cn);
}

// ---------------------------------------------------------------------------
// x0 = ind_f @ pw0 + o @ pw1 + pb1 ; optionally emit as output column outIdx.
// ---------------------------------------------------------------------------
__global__ void x0_kernel(const float* __restrict__ indf,
                          const float* __restrict__ o,
                          const float* __restrict__ pw0,
                          const float* __restrict__ pw1,
                          const float* __restrict__ pb1,
                          float* __restrict__ x0,
                          float* __restrict__ out, int outIdx) {
    int i = blockIdx.x * blockDim.x + threadIdx.x;
    if (i >= NN) return;
    float v = pb1[0];
    #pragma unroll
    for (int f = 0; f < NIN0; ++f) v = fmaf(indf[(size_t)i*NIN0 + f], pw0[f], v);
    #pragma unroll
    for (int j = 0; j < NF1; ++j)  v = fmaf(o[(size_t)i*NF1 + j], pw1[j], v);
    x0[i] = v;
    if (outIdx >= 0) out[(size_t)i * 3 + outIdx] = v;
}

// ---------------------------------------------------------------------------
extern "C" void kernel_launch(void* const* d_in, const int* in_sizes, int n_in,
                              void* d_out, int out_size, void* d_ws, size_t ws_size,
                              hipStream_t stream) {
    const int*   species  = (const int*)  d_in[0];
    const float* coords   = (const float*)d_in[1];
    const float* x_h      = (const float*)d_in[2];
    const float* x_raw    = (const float*)d_in[3];
    const int*   pf       = (const int*)  d_in[4];
    const int*   ps       = (const int*)  d_in[5];
    const float* h0_int_w = (const float*)d_in[6];   // (20,5,40)  -> (100,40)
    const float* h0_self_w= (const float*)d_in[7];   // (5,40)
    const float* h0_self_b= (const float*)d_in[8];   // (40)
    const float* h0_aw    = (const float*)d_in[9];   // (3,40,40)
    const float* h0_ab    = (const float*)d_in[10];  // (3,40)
    const float* h1_int_w = (const float*)d_in[11];  // (20,1,20)  -> (20,20)
    const float* h1_self_w= (const float*)d_in[12];  // (1,20)
    const float* h1_self_b= (const float*)d_in[13];  // (20)
    const float* h1_aw    = (const float*)d_in[14];  // (3,20,20)
    const float* h1_ab    = (const float*)d_in[15];  // (3,20)
    const float* W_int_w  = (const float*)d_in[16];  // (20,40,80) -> (800,80)
    const float* W_self_w = (const float*)d_in[17];  // (40,80)
    const float* W_self_b = (const float*)d_in[18];  // (80)
    const float* pw0      = (const float*)d_in[19];  // (5,1)
    const float* pw1      = (const float*)d_in[20];  // (20,1)
    const float* pb1      = (const float*)d_in[21];  // (1)
    float* out = (float*)d_out;                      // (N,3)

    // Workspace carve (all 256B-aligned chunks), ~30 MB total, fits L2.
    float* wsp = (float*)d_ws;
    size_t off = 0;
    auto carve = [&](size_t n) { float* p = wsp + off; off += (n + 63) & ~(size_t)63; return p; };
    float* sense = carve((size_t)PP * NSENS);    // 10.5 MB
    float* indf  = carve((size_t)NN * NIN0);
    float* env0  = carve((size_t)NN * NSENS * NIN0);
    float* ya    = carve((size_t)NN * NF0);
    float* yb    = carve((size_t)NN * NF0);
    float* cat   = carve((size_t)NN * NF0);      // [x_t | h_t]
    float* cbuf  = carve((size_t)NN * NF1);
    float* obuf  = carve((size_t)NN * NF1);
    float* env1  = carve((size_t)NN * NSENS);
    float* y1a   = carve((size_t)NN * NF1);
    float* y1b   = carve((size_t)NN * NF1);
    float* envW  = carve((size_t)NN * NSENS * NF0); // 13.1 MB
    float* tmp80 = carve((size_t)NN * NG);
    float* x0    = carve((size_t)NN);

    auto gemm = [&](int act, const float* A, int lda, int K,
                    const float* Wm, int ldw,
                    const float* X, int xs, int KX, const float* S,
                    const float* b, float* Dp, int ldd, int dcol, int Nc) {
        dim3 grid(NN / 64, (Nc + 15) / 16);
        dim3 blk(32, 4);
        if (act)
            gemm_wmma<1><<<grid, blk, 0, stream>>>(A, lda, K, Wm, ldw, X, xs, KX, S, b, Dp, ldd, dcol, Nc);
        else
            gemm_wmma<0><<<grid, blk, 0, stream>>>(A, lda, K, Wm, ldw, X, xs, KX, S, b, Dp, ldd, dcol, Nc);
    };

    // ---- geometry / features ----
    sense_kernel<<<PP / 256, 256, 0, stream>>>(coords, pf, ps, sense);
    indf_kernel <<<NN / 256, 256, 0, stream>>>(species, x_h, indf);

    // ---- h0 HIPNN block -> f0_out (N,40) ----
    env_kernel<NIN0><<<NN, 128, 0, stream>>>(indf, NIN0, ps, sense, env0);
    gemm(1, env0, NSENS*NIN0, NSENS*NIN0, h0_int_w, NF0,
         indf, NIN0, NIN0, h0_self_w, h0_self_b, ya, NF0, 0, NF0);
    gemm(1, ya, NF0, NF0, h0_aw + 0*NF0*NF0, NF0, nullptr, 0, 0, nullptr, h0_ab + 0*NF0, yb, NF0, 0, NF0);
    gemm(1, yb, NF0, NF0, h0_aw + 1*NF0*NF0, NF0, nullptr, 0, 0, nullptr, h0_ab + 1*NF0, ya, NF0, 0, NF0);
    gemm(1, ya, NF0, NF0, h0_aw + 2*NF0*NF0, NF0, nullptr, 0, 0, nullptr, h0_ab + 2*NF0, yb, NF0, 0, NF0);
    split_kernel<<<(NN*NF1 + 255) / 256, 256, 0, stream>>>(yb, cbuf, cat);

    // ---- 5 LSTM steps: 2 burn-in (x_raw cols), 3 output (x0 feedback) ----
    for (int step = 0; step < 5; ++step) {
        const float* xin; int xs;
        if      (step == 0) { xin = x_raw;     xs = 2; }
        else if (step == 1) { xin = x_raw + 1; xs = 2; }
        else                { xin = x0;        xs = 1; }

        // x_t = hipnn1(x_in): interact (env1 @ 20x20 + self) + 3 atom layers
        env_kernel<1><<<NN, 128, 0, stream>>>(xin, xs, ps, sense, env1);
        gemm(1, env1, NSENS, NSENS, h1_int_w, NF1, xin, xs, 1, h1_self_w, h1_self_b, y1a, NF1, 0, NF1);
        gemm(1, y1a, NF1, NF1, h1_aw + 0*NF1*NF1, NF1, nullptr, 0, 0, nullptr, h1_ab + 0*NF1, y1b, NF1, 0, NF1);
        gemm(1, y1b, NF1, NF1, h1_aw + 1*NF1*NF1, NF1, nullptr, 0, 0, nullptr, h1_ab + 1*NF1, y1a, NF1, 0, NF1);
        gemm(1, y1a, NF1, NF1, h1_aw + 2*NF1*NF1, NF1, nullptr, 0, 0, nullptr, h1_ab + 2*NF1, cat, NF0, 0, NF1);

        // tmp = interact([x_t|h_t]) : env(4096x800) @ (800x80) + cat @ (40x80) + b
        env_kernel<NF0><<<NN, 128, 0, stream>>>(cat, NF0, ps, sense, envW);
        gemm(0, envW, NSENS*NF0, NSENS*NF0, W_int_w, NG, cat, NF0, NF0, W_self_w, W_self_b, tmp80, NG, 0, NG);

        gate_kernel<<<(NN*NF1 + 255) / 256, 256, 0, stream>>>(tmp80, cbuf, obuf, cat);

        // x0 recompute after 2nd..5th steps; steps 2..4 emit output cols 0..2
        if (step >= 1)
            x0_kernel<<<(NN + 255) / 256, 256, 0, stream>>>(indf, obuf, pw0, pw1, pb1, x0, out, step - 2);
    }
}